// Model_wat_14817637171534
// MI455X (gfx1250) — compile-verified
//
#include <hip/hip_runtime.h>

// CDNA5 / gfx1250: wave32, WMMA 16x16x4 f32.
typedef float v2f __attribute__((ext_vector_type(2)));
typedef float v8f __attribute__((ext_vector_type(8)));

#define NG      48
#define NVOX    (NG * NG * NG)        // 110592
#define N_ATOMS 1024
#define NTILE   (N_ATOMS / 16)        // 64 atom tiles
#define GRIDSP  0.5f
#define R2      5.1984f               // (1.5 * 1.52)^2
#define WEIGHT  25.0f
#define BLOCK   256                   // 8 wave32 waves
#define VOX_PER_BLOCK (BLOCK / 32 * 16)   // 128

__global__ __launch_bounds__(BLOCK) void wat_count_kernel(
    const float* __restrict__ vecs,   // [N_ATOMS, 3] f32
    float* __restrict__ out)          // [2, NG, NG, NG] f32
{
    // Atom data staged in LDS directly in WMMA A-operand layout:
    //   sA[tile*32 + m]      = (x, y)        -> lanes 0..15  (K=0,1), M=m
    //   sA[tile*32 + 16 + m] = (z, |a|^2)    -> lanes 16..31 (K=2,3), M=m
    __shared__ v2f sA[NTILE * 32];    // 16 KB
    for (int i = threadIdx.x; i < N_ATOMS; i += BLOCK) {
        float x = vecs[3 * i + 0];
        float y = vecs[3 * i + 1];
        float z = vecs[3 * i + 2];
        int tile = i >> 4, m = i & 15;
        sA[tile * 32 + m]      = (v2f){x, y};
        sA[tile * 32 + 16 + m] = (v2f){z, x * x + y * y + z * z};
    }
    __syncthreads();

    const int lane  = threadIdx.x & 31;
    const int wave  = threadIdx.x >> 5;       // 0..7
    const int laneN = lane & 15;              // voxel column within tile
    const bool hi   = lane >= 16;

    // This wave's 16 voxels (864 blocks * 128 = NVOX exactly, no tail).
    const int v  = (blockIdx.x * 8 + wave) * 16 + laneN;
    const int kz = v % NG;
    const int t  = v / NG;
    const int jy = t % NG;
    const int ix = t / NG;
    const float gx = GRIDSP * (float)ix;
    const float gy = GRIDSP * (float)jy;
    const float gz = GRIDSP * (float)kz;
    const float gsq = gx * gx + gy * gy + gz * gz;

    // B (4x16): col n = (-2gx, -2gy, -2gz, 1). Lanes 0-15: K0,K1; lanes 16-31: K2,K3.
    v2f B;
    B.x = hi ? (-2.0f * gz) : (-2.0f * gx);
    B.y = hi ? 1.0f         : (-2.0f * gy);

    const v8f Czero = {};               // lowers to inline 0 accumulator

    // D[m][n] = |a_m|^2 - 2 a_m.g_n ; voxel n is "solvent-on" iff
    // min_m D[m][n] < R2 - |g_n|^2  (counts >= 0.9 <=> count >= 1 <=> min d^2 < r^2).
    const float thresh = R2 - gsq;
    float dmin = 3.0e38f;

#pragma unroll 4
    for (int tile = 0; tile < NTILE; ++tile) {
        v2f A = sA[tile * 32 + lane];   // ds_load_b64, already in A layout
        v8f D = __builtin_amdgcn_wmma_f32_16x16x4_f32(
            /*neg_a=*/false, A, /*neg_b=*/false, B,
            /*c_mod=*/(short)0, Czero, /*reuse_a=*/false, /*reuse_b=*/false);
        // 8-wide min tree (compiler can fuse into v_min3_f32).
        float m01 = fminf(D[0], D[1]);
        float m23 = fminf(D[2], D[3]);
        float m45 = fminf(D[4], D[5]);
        float m67 = fminf(D[6], D[7]);
        float m0123 = fminf(m01, m23);
        float m4567 = fminf(m45, m67);
        dmin = fminf(dmin, fminf(m0123, m4567));
    }

    // Lane L holds rows M=0..7, lane L+16 rows M=8..15 of the same voxel column.
    dmin = fminf(dmin, __shfl_xor(dmin, 16, 32));

    if (!hi) {
        const bool cond = (dmin < thresh);
        out[v]        = cond ? 1.0f   : 0.0f;     // channel 0
        out[NVOX + v] = cond ? WEIGHT : 1.0f;     // channel 1
    }
}

extern "C" void kernel_launch(void* const* d_in, const int* in_sizes, int n_in,
                              void* d_out, int out_size, void* d_ws, size_t ws_size,
                              hipStream_t stream) {
    (void)in_sizes; (void)n_in; (void)d_ws; (void)ws_size; (void)out_size;
    const float* vecs = (const float*)d_in[0];   // [1024,3] f32
    float* out = (float*)d_out;                  // [2,48,48,48] f32
    const int blocks = NVOX / VOX_PER_BLOCK;     // 864, exact
    wat_count_kernel<<<blocks, BLOCK, 0, stream>>>(vecs, out);
}